// MSDeformAttn_73126113181735
// MI455X (gfx1250) — compile-verified
//
#include <hip/hip_runtime.h>
#include <hip/hip_bf16.h>
#include <math.h>

// MS-Deformable-Attention for gfx1250 (MI455X), f32 end-to-end.
// GEMMs: V_WMMA_F32_16X16X4_F32 with the shared W-tile staged into LDS by the
// Tensor Data Mover (tensor_load_to_lds + s_wait_tensorcnt), using TDM row
// padding (64+8 dwords) so all B-fragment ds reads are bank-conflict-free.

typedef __attribute__((ext_vector_type(2))) float    v2f;
typedef __attribute__((ext_vector_type(8))) float    v8f;
typedef __attribute__((ext_vector_type(4))) unsigned v4u;
typedef __attribute__((ext_vector_type(8))) int      v8i;
typedef __attribute__((ext_vector_type(4))) int      v4i;

#define BS      2
#define QLEN    21760
#define LEN_IN  21760
#define DMODEL  256
#define NHEADS  8
#define HDIM    32
#define NLVL    4
#define NPTS    4

#define LDS_ROW 72   // 64 data dwords + 8 pad dwords per tile row (bank swizzle)

__constant__ const int LVL_H[NLVL]     = {128, 64, 32, 16};
__constant__ const int LVL_W[NLVL]     = {128, 64, 32, 16};
__constant__ const int LVL_START[NLVL] = {0, 16384, 20480, 21504};

// ---------------------------------------------------------------------------
// TDM: load a 2D tile (256 rows x 64 contiguous cols, row stride = N floats)
// from global into LDS, one padded row per K.  D# packed per ISA 8.3/8.4:
//   G0: count=1 | lds_addr | global_addr(57b) | type=2
//   G1: wg_mask=0, data_size=2(4B), pad_enable=1, pad_interval=5 (64 dwords),
//       pad_amount=7 (8 dwords), tensor_dim0=N, tensor_dim1=256,
//       tile_dim0=64, tile_dim1=256, tensor_dim0_stride=N
//   G2/G3: zero (<=2D tensor)
// Resulting LDS layout: row k at byte (k*LDS_ROW*4); bank(k,c) = (8k+c)%64,
// so the 16-lane halves of a wave (K-pairs k and k+2) hit disjoint bank sets.
// ---------------------------------------------------------------------------
__device__ __forceinline__ void tdm_load_w_tile(const float* gptr,
                                                unsigned lds_byte_off, int N) {
  const unsigned long long ga = (unsigned long long)(const void*)gptr;
  v4u g0;
  g0[0] = 1u;                                            // count=1, user mode
  g0[1] = lds_byte_off;                                  // LDS byte address
  g0[2] = (unsigned)(ga & 0xFFFFFFFFu);                  // global_addr[31:0]
  g0[3] = (unsigned)((ga >> 32) & 0x01FFFFFFu) | (2u << 30);  // [56:32]|type=2
  v8i g1;
  const unsigned uN = (unsigned)N;
  g1[0] = (int)((2u << 16)        // data_size = 4 bytes
              | (1u << 20)        // pad_enable
              | (5u << 22)        // pad_interval: every 64 dwords
              | (7u << 25));      // pad_amount: 8 dwords
  g1[1] = (int)((uN & 0xFFFFu) << 16);                   // tensor_dim0 lo16
  g1[2] = (int)(((uN >> 16) & 0xFFFFu) | (256u << 16));  // dim0 hi | dim1 lo
  g1[3] = (int)(64u << 16);                              // dim1 hi=0 | tile_dim0=64
  g1[4] = (int)256u;                                     // tile_dim1=256, tile_dim2=0
  g1[5] = (int)uN;                                       // tensor_dim0_stride lo32
  g1[6] = 0;                                             // stride hi | dim1_stride lo
  g1[7] = 0;
  const v4i z4 = {0, 0, 0, 0};
  const v8i z8 = {0, 0, 0, 0, 0, 0, 0, 0};
  __builtin_amdgcn_tensor_load_to_lds(g0, g1, z4, z4, z8, 0);
}

// ---------------------------------------------------------------------------
// C[M,N] = A[M,256] @ W[256,N] + bias[N]
// grid = (N/64, M/128), block = 256 threads (8 waves).
// Wave w: rows [by*128 + w*16, +16), cols [bx*64, +64) as 4 WMMA n-chunks.
// W tile (72KB padded) staged once into LDS by TDM (wave 0), shared by 8 waves.
//
// f32 WMMA fragment layouts (ISA 7.12.2):
//   A 16x4:  lane(0..15)=M, VGPR0 = K=(2*hi), VGPR1 = K=(2*hi+1)   [hi=lane>>4]
//   B 4x16:  lane(0..15)=N, VGPR0 = K=(2*hi), VGPR1 = K=(2*hi+1)
//   C/D 16x16: VGPR r holds M=r (lanes 0-15) / M=8+r (lanes 16-31), N=lane&15
// ---------------------------------------------------------------------------
__global__ __launch_bounds__(256)
void gemm_k256_wmma(const float* __restrict__ A, const float* __restrict__ W,
                    const float* __restrict__ bias, float* __restrict__ C,
                    int N) {
  __shared__ float wlds[256 * LDS_ROW];   // padded [k][c] tile, 72 KB

  const int K    = 256;
  const int lane = threadIdx.x & 31;
  const int wave = threadIdx.x >> 5;
  const int lo   = lane & 15;
  const int hi   = lane >> 4;           // 0 or 1
  const int m0   = blockIdx.y * 128 + wave * 16;
  const int n0   = blockIdx.x * 64;

  // Wave 0 DMAs the W tile into LDS; everyone syncs before reading it.
  if (threadIdx.x < 32) {
    tdm_load_w_tile(W + n0, (unsigned)(unsigned long long)(void*)&wlds[0], N);
    __builtin_amdgcn_s_wait_tensorcnt(0);
  }
  __syncthreads();

  v8f acc[4];
#pragma unroll
  for (int nc = 0; nc < 4; ++nc) {
    const float bv = bias[n0 + nc * 16 + lo];  // column bias, same for all rows
#pragma unroll
    for (int r = 0; r < 8; ++r) acc[nc][r] = bv;
  }

  const float* arow = A + (long)(m0 + lo) * K;

#pragma unroll 4
  for (int k = 0; k < K; k += 4) {
    // A fragment: contiguous float2 per lane (global, coalesced b64).
    const v2f a = *(const v2f*)(arow + k + 2 * hi);
    const int kr = (k + 2 * hi) * LDS_ROW;  // padded LDS row for this K pair
#pragma unroll
    for (int nc = 0; nc < 4; ++nc) {
      const int c = nc * 16 + lo;
      v2f b;
      b.x = wlds[kr + c];                   // W[k+2hi   ][n0+c]
      b.y = wlds[kr + LDS_ROW + c];         // W[k+2hi+1 ][n0+c]
      acc[nc] = __builtin_amdgcn_wmma_f32_16x16x4_f32(
          /*neg_a=*/false, a, /*neg_b=*/false, b,
          /*c_mod=*/(short)0, acc[nc], /*reuse_a=*/false, /*reuse_b=*/false);
    }
  }

#pragma unroll
  for (int nc = 0; nc < 4; ++nc) {
    const int col = n0 + nc * 16 + lo;
#pragma unroll
    for (int r = 0; r < 8; ++r) {
      C[(long)(m0 + r + 8 * hi) * N + col] = acc[nc][r];
    }
  }
}

// ---------------------------------------------------------------------------
// In-place softmax over the trailing 16 (levels*points) per (b,q,h).
// ---------------------------------------------------------------------------
__global__ __launch_bounds__(256)
void softmax16_kernel(float* __restrict__ attn, int total) {
  const int i = blockIdx.x * blockDim.x + threadIdx.x;
  if (i >= total) return;
  float* p = attn + (long)i * 16;
  float e[16];
  float m = p[0];
#pragma unroll
  for (int j = 1; j < 16; ++j) m = fmaxf(m, p[j]);
  float s = 0.f;
#pragma unroll
  for (int j = 0; j < 16; ++j) { e[j] = __expf(p[j] - m); s += e[j]; }
  const float inv = 1.f / s;
#pragma unroll
  for (int j = 0; j < 16; ++j) p[j] = e[j] * inv;
}

// ---------------------------------------------------------------------------
// Deformable bilinear sampling. One wave32 per (b,q,h); lane = channel d.
// sampled[(b*Q+q)*256 + h*32 + d] = sum_{l,p} attn * bilinear(value, loc)
// Corner gathers are 32-lane contiguous (128B) reads of value.
// ---------------------------------------------------------------------------
__global__ __launch_bounds__(256)
void ms_deform_sample(const float* __restrict__ value,
                      const float* __restrict__ refpts,
                      const float* __restrict__ off,
                      const float* __restrict__ attn,
                      float* __restrict__ sampled) {
  const int lane = threadIdx.x & 31;          // channel within head
  const int wave = threadIdx.x >> 5;
  const int wid  = blockIdx.x * 8 + wave;     // (b*Q + q)*8 + h
  const int h    = wid & 7;
  const int bq   = wid >> 3;                  // b*Q + q
  const int b    = bq / QLEN;

  const long off_base  = (long)bq * (NHEADS * NLVL * NPTS * 2);
  const long attn_base = (long)bq * (NHEADS * NLVL * NPTS);
  const long ref_base  = (long)bq * (NLVL * 2);
  const long vbase     = (long)b * LEN_IN;    // row offset into value
  const int  ch        = h * HDIM + lane;     // column in value/sampled

  float acc = 0.f;

#pragma unroll
  for (int l = 0; l < NLVL; ++l) {
    const int H = LVL_H[l], W = LVL_W[l], st = LVL_START[l];
    const float rx = refpts[ref_base + l * 2 + 0];
    const float ry = refpts[ref_base + l * 2 + 1];
#pragma unroll
    for (int p = 0; p < NPTS; ++p) {
      const long oi = off_base + (((h * NLVL + l) * NPTS + p) << 1);
      const float ox = off[oi], oy = off[oi + 1];
      const float aw = attn[attn_base + (h * NLVL + l) * NPTS + p];

      // loc = ref + off/(W,H); grid=2*loc-1; x = ((gx+1)*W-1)/2 = rx*W+ox-0.5
      const float x = rx * (float)W + ox - 0.5f;
      const float y = ry * (float)H + oy - 0.5f;
      const float x0f = floorf(x), y0f = floorf(y);
      const float wx = x - x0f, wy = y - y0f;
      const int x0 = (int)x0f, y0 = (int)y0f;
      const int x1 = x0 + 1, y1 = y0 + 1;

      const bool vx0 = (x0 >= 0) & (x0 < W);
      const bool vx1 = (x1 >= 0) & (x1 < W);
      const bool vy0 = (y0 >= 0) & (y0 < H);
      const bool vy1 = (y1 >= 0) & (y1 < H);

      float v00 = 0.f, v01 = 0.f, v10 = 0.f, v11 = 0.f;
      if (vy0 & vx0) v00 = value[(vbase + st + (long)y0 * W + x0) * DMODEL + ch];
      if (vy0 & vx1) v01 = value[(vbase + st + (long)y0 * W + x1) * DMODEL + ch];
      if (vy1 & vx0) v10 = value[(vbase + st + (long)y1 * W + x0) * DMODEL + ch];
      if (vy1 & vx1) v11 = value[(vbase + st + (long)y1 * W + x1) * DMODEL + ch];

      const float bil = v00 * (1.f - wy) * (1.f - wx) + v01 * (1.f - wy) * wx +
                        v10 * wy * (1.f - wx)         + v11 * wy * wx;
      acc += aw * bil;
    }
  }

  sampled[(long)bq * DMODEL + ch] = acc;
}

// ---------------------------------------------------------------------------
extern "C" void kernel_launch(void* const* d_in, const int* in_sizes, int n_in,
                              void* d_out, int out_size, void* d_ws, size_t ws_size,
                              hipStream_t stream) {
  const float* query   = (const float*)d_in[0];
  const float* refpts  = (const float*)d_in[1];
  const float* x_flat  = (const float*)d_in[2];
  const float* W_value = (const float*)d_in[3];
  const float* b_value = (const float*)d_in[4];
  const float* W_off   = (const float*)d_in[5];
  const float* b_off   = (const float*)d_in[6];
  const float* W_attn  = (const float*)d_in[7];
  const float* b_attn  = (const float*)d_in[8];
  const float* W_out   = (const float*)d_in[9];
  const float* b_out   = (const float*)d_in[10];
  float* out = (float*)d_out;

  const int M = BS * QLEN;  // 43520, divisible by 128

  float* ws      = (float*)d_ws;
  float* value   = ws;                              // M * 256
  float* off     = value   + (long)M * DMODEL;      // M * 256
  float* attn    = off     + (long)M * DMODEL;      // M * 128
  float* sampled = attn    + (long)M * 128;         // M * 256

  const dim3 blk(256);

  // 1) value = input_flatten @ W_value + b_value   (N=256)
  gemm_k256_wmma<<<dim3(256 / 64, M / 128), blk, 0, stream>>>(
      x_flat, W_value, b_value, value, 256);

  // 2) off = query @ W_off + b_off                 (N=256)
  gemm_k256_wmma<<<dim3(256 / 64, M / 128), blk, 0, stream>>>(
      query, W_off, b_off, off, 256);

  // 3) attn logits = query @ W_attn + b_attn       (N=128)
  gemm_k256_wmma<<<dim3(128 / 64, M / 128), blk, 0, stream>>>(
      query, W_attn, b_attn, attn, 128);

  // 4) softmax over 16 per (b,q,h)
  const int nsm = M * NHEADS;  // 348160
  softmax16_kernel<<<(nsm + 255) / 256, blk, 0, stream>>>(attn, nsm);

  // 5) deformable bilinear sampling: one wave per (b,q,h)
  ms_deform_sample<<<nsm / 8, blk, 0, stream>>>(value, refpts, off, attn, sampled);

  // 6) out = sampled @ W_out + b_out               (N=256)
  gemm_k256_wmma<<<dim3(256 / 64, M / 128), blk, 0, stream>>>(
      sampled, W_out, b_out, out, 256);
}